// E76LogSpaceDeltaCell_83829171683352
// MI455X (gfx1250) — compile-verified
//
#include <hip/hip_runtime.h>
#include <hip/hip_bf16.h>
#include <math.h>

// ---------------------------------------------------------------------------
// E76 log-space delta cell for MI455X (gfx1250, wave32).
//   T=2048, B=8, D=1024, n=64
// Kernel 1: fused k/v/q/gate projections via V_WMMA_F32_16X16X4_F32 (exact f32).
// Kernel 2: sequential scan, 1 block per batch, 256 threads (row, quarter),
//           state rows held in VGPRs, 4-lane shuffle reductions.
// ---------------------------------------------------------------------------

typedef __attribute__((ext_vector_type(2))) float v2f;
typedef __attribute__((ext_vector_type(8))) float v8f;

#define T_DIM 2048
#define B_DIM 8
#define D_DIM 1024
#define N_DIM 64
#define TBN ((size_t)T_DIM * B_DIM * N_DIM)

// ---------------------------------------------------------------------------
// Projection: C[r, c] = sum_d x[r, d] * W[c, d]   (r = t*B+b flat row)
// One 16x16 C tile per wave; K stepped by 4 via f32 WMMA.
// A frag (16x4 f32): lanes 0-15 -> M=lane, K={0,1}; lanes 16-31 -> M=lane-16,
// K={2,3}  => one float2 load per lane per WMMA. B frag mirrors with W^T.
// C frag (8 VGPRs): VGPR r holds rows {r, r+8}, N = lane&15.
// ---------------------------------------------------------------------------
__global__ __launch_bounds__(256) void e76_proj_wmma(
    const float* __restrict__ x,
    const float* __restrict__ Wk, const float* __restrict__ Wv,
    const float* __restrict__ Wq, const float* __restrict__ Wg,
    float* __restrict__ ok, float* __restrict__ ov,
    float* __restrict__ oq, float* __restrict__ og)
{
    const int lane = threadIdx.x & 31;
    const int wave = threadIdx.x >> 5;
    const int w    = blockIdx.x * 8 + wave;   // global wave id
    const int mt   = w >> 4;                  // row tile   (0..1023)
    const int ctg  = w & 15;                  // col group  (0..15): 4 per weight

    const float* Wp;
    float* op;
    switch (ctg >> 2) {
        case 0:  Wp = Wk; op = ok; break;
        case 1:  Wp = Wv; op = ov; break;
        case 2:  Wp = Wq; op = oq; break;
        default: Wp = Wg; op = og; break;
    }
    const int col  = (ctg & 3) * 16;
    const int lrow = lane & 15;
    const int koff = (lane >> 4) * 2;         // 0 for lanes 0-15, 2 for 16-31

    const float* aptr = x  + (size_t)(mt * 16 + lrow) * D_DIM + koff;
    const float* bptr = Wp + (size_t)(col + lrow)     * D_DIM + koff;

    v8f c = {};                               // f32 accumulator, 8 VGPRs
    #pragma unroll 1
    for (int k0 = 0; k0 < D_DIM; k0 += 8) {
        v2f a0 = *(const v2f*)(aptr + k0);
        v2f b0 = *(const v2f*)(bptr + k0);
        v2f a1 = *(const v2f*)(aptr + k0 + 4);
        v2f b1 = *(const v2f*)(bptr + k0 + 4);
        c = __builtin_amdgcn_wmma_f32_16x16x4_f32(false, a0, false, b0,
                                                  (short)0, c, false, false);
        c = __builtin_amdgcn_wmma_f32_16x16x4_f32(false, a1, false, b1,
                                                  (short)0, c, false, false);
    }

    const int rbase = mt * 16 + ((lane >> 4) ? 8 : 0);
    const int cg    = col + lrow;
    #pragma unroll
    for (int r = 0; r < 8; ++r)
        op[(size_t)(rbase + r) * N_DIM + cg] = c[r];
}

// ---------------------------------------------------------------------------
// Sequential scan. Block b handles batch b. 256 threads:
//   row i = tid >> 2  (0..63),  quarter qq = tid & 3 (columns qq*16..qq*16+15)
// Each thread keeps 16 state values in VGPRs. 4-lane dot reductions via
// __shfl_xor (same wave). ||k||^2 via LDS.
// ---------------------------------------------------------------------------
__device__ __forceinline__ float red4(float v) {
    v += __shfl_xor(v, 1);
    v += __shfl_xor(v, 2);
    return v;
}

__global__ __launch_bounds__(256) void e76_scan(
    const float* __restrict__ kbuf, const float* __restrict__ vbuf,
    const float* __restrict__ qbuf, const float* __restrict__ gbuf,
    const float* __restrict__ S0,   const float* __restrict__ A_log,
    const float* __restrict__ dt_bias,
    float* __restrict__ out, float* __restrict__ Sf)
{
    const int b   = blockIdx.x;
    const int tid = threadIdx.x;
    const int row = tid >> 2;            // 0..63
    const int qq  = tid & 3;             // 0..3
    const int c0  = qq * 16;             // first owned column

    __shared__ float ksh[N_DIM];
    __shared__ float qsh[N_DIM];
    __shared__ float red[N_DIM];

    // state row slice in registers (fully unrolled accesses only)
    float s[16];
    {
        const float* src = S0 + ((size_t)b * N_DIM + row) * N_DIM + c0;
        #pragma unroll
        for (int j = 0; j < 16; ++j) s[j] = src[j];
    }

    const float Ai  = __expf(A_log[row]);
    const float dtb = dt_bias[row];

    #pragma unroll 1
    for (int t = 0; t < T_DIM; ++t) {
        const size_t base = ((size_t)t * B_DIM + b) * N_DIM;

        // stage k, q, k^2 into LDS (first 64 threads)
        if (tid < N_DIM) {
            float kv = kbuf[base + tid];
            ksh[tid] = kv;
            red[tid] = kv * kv;
            qsh[tid] = qbuf[base + tid];
        }
        const float vv = vbuf[base + row];
        const float gg = gbuf[base + row];
        __syncthreads();

        // ||k|| (every thread sums 64 broadcast LDS reads)
        float ss = 0.f;
        #pragma unroll
        for (int j = 0; j < N_DIM; ++j) ss += red[j];
        const float inv = 1.0f / (sqrtf(ss) + 1e-6f);

        // decay = exp(-A * softplus(gate + dt_bias))
        const float z     = gg + dtb;
        const float dt    = (z > 20.f) ? z : log1pf(__expf(z));
        const float decay = __expf(-Ai * dt);

        // retrieved[row] = inv * sum_j S[row,j] * k[j]
        float rp = 0.f;
        #pragma unroll
        for (int j = 0; j < 16; ++j) rp = fmaf(s[j], ksh[c0 + j], rp);
        const float delta = vv - red4(rp) * inv;

        // S = tanh(decay*S + delta (x) k_norm);  Sq[row] = sum_j S[row,j]*q[j]
        float sqp = 0.f;
        #pragma unroll
        for (int j = 0; j < 16; ++j) {
            const float kn = ksh[c0 + j] * inv;
            const float a  = fmaf(decay, s[j], delta * kn);
            const float e  = __expf(2.0f * a);           // tanh via exp
            const float v2 = 1.0f - 2.0f / (e + 1.0f);   // -> ±1 at ±inf
            s[j] = v2;
            sqp  = fmaf(v2, qsh[c0 + j], sqp);
        }
        const float sq = red4(sqp);
        if (qq == 0) {
            const float sig = 1.0f / (1.0f + __expf(-sq));
            out[base + row] = sq * sq * sig;             // Sq * silu(Sq)
        }
        __syncthreads();   // protect LDS before next step's staging
    }

    // final state
    {
        float* dst = Sf + ((size_t)b * N_DIM + row) * N_DIM + c0;
        #pragma unroll
        for (int j = 0; j < 16; ++j) dst[j] = s[j];
    }
}

// ---------------------------------------------------------------------------
extern "C" void kernel_launch(void* const* d_in, const int* in_sizes, int n_in,
                              void* d_out, int out_size, void* d_ws, size_t ws_size,
                              hipStream_t stream) {
    const float* x       = (const float*)d_in[0];
    const float* S0      = (const float*)d_in[1];
    const float* Wk      = (const float*)d_in[2];
    const float* Wv      = (const float*)d_in[3];
    const float* Wq      = (const float*)d_in[4];
    const float* Wg      = (const float*)d_in[5];
    const float* A_log   = (const float*)d_in[6];
    const float* dt_bias = (const float*)d_in[7];

    float* wk = (float*)d_ws;        // 4 projection buffers, 4 MB each
    float* wv = wk + TBN;
    float* wq = wv + TBN;
    float* wg = wq + TBN;

    float* out = (float*)d_out;      // [T, B, n]
    float* Sf  = out + TBN;          // [B, n, n]

    // 1024 row-tiles * 16 col-groups = 16384 wave-tiles / 8 waves per block
    e76_proj_wmma<<<2048, 256, 0, stream>>>(x, Wk, Wv, Wq, Wg, wk, wv, wq, wg);
    e76_scan<<<B_DIM, 256, 0, stream>>>(wk, wv, wq, wg, S0, A_log, dt_bias,
                                        out, Sf);
}